// FcosPostProcess_16733192585468
// MI455X (gfx1250) — compile-verified
//
#include <hip/hip_runtime.h>
#include <float.h>

typedef __attribute__((ext_vector_type(2))) float v2f;
typedef __attribute__((ext_vector_type(8))) float v8f;

#define NIMG        32
#define NCH         85
#define NANCH       13343
#define ANCH_PAD    13344
#define NSORT       16384
#define K_PRE       1000
#define MAX_DET     300
#define TILES_PER_IMG 837
#define WAVES_PER_BLK 8

__device__ __forceinline__ float sigm(float x) { return 1.0f / (1.0f + expf(-x)); }

// ---------------------------------------------------------------------------
// Kernel 1: fused decode + score. One wave32 handles a 16-anchor tile of one
// level. Box decode (the only linear-algebra part) goes through
// V_WMMA_F32_16X16X4_F32: D(16x16) = A(16x4: raw box preds) x T(4x16: diag
// +-stride) ; anchor centers added afterwards (exact: strides are pow2).
// ---------------------------------------------------------------------------
__global__ __launch_bounds__(256) void decode_kernel(
    const float* __restrict__ p0, const float* __restrict__ p1,
    const float* __restrict__ p2, const float* __restrict__ p3,
    const float* __restrict__ p4,
    float* __restrict__ wsBox, float* __restrict__ wsCls,
    float* __restrict__ wsScore)
{
    const int LSIZE[5] = {10000, 2500, 625, 169, 49};
    const int LNX[5]   = {100, 50, 25, 13, 7};
    const float LSTR[5]= {8.0f, 16.0f, 32.0f, 64.0f, 128.0f};
    const int LOFF[5]  = {0, 10000, 12500, 13125, 13294};
    const int LTC[6]   = {0, 625, 782, 822, 833, 837};

    __shared__ float shD[WAVES_PER_BLK][4][16];

    const int wave = threadIdx.x >> 5;
    const int lane = threadIdx.x & 31;
    const int tile = blockIdx.x * WAVES_PER_BLK + wave;
    const int img  = tile / TILES_PER_IMG;
    const int t    = tile - img * TILES_PER_IMG;

    int lvl = 0;
    while (lvl < 4 && t >= LTC[lvl + 1]) lvl++;
    const int a0 = (t - LTC[lvl]) * 16;
    const float* pr = (lvl == 0) ? p0 : (lvl == 1) ? p1 : (lvl == 2) ? p2
                       : (lvl == 3) ? p3 : p4;
    const int   nx = LNX[lvl];
    const int   LH = LSIZE[lvl];       // ny*nx (channel stride)
    const float s  = LSTR[lvl];

    const int  m     = lane & 15;      // anchor-in-tile (row M)
    const bool hi    = lane >= 16;
    const int  a     = a0 + m;         // anchor index within level
    const bool valid = a < LH;
    const size_t base = (size_t)img * NCH * (size_t)LH + (size_t)a;

    // A fragment (32-bit 16x4): lanes 0-15 -> K=0,1 ; lanes 16-31 -> K=2,3
    const int c0 = hi ? 2 : 0;
    v2f af;
    af.x = valid ? pr[base + (size_t)c0 * LH]       : 0.0f;
    af.y = valid ? pr[base + (size_t)(c0 + 1) * LH] : 0.0f;

    // B fragment (4x16): v0 = rows K=0 (lanes 0-15) / K=2 (lanes 16-31),
    //                    v1 = rows K=1 / K=3.  T = diag(-s,-s,s,s) in cols 0..3.
    v2f bf;
    bf.x = (lane == 0) ? -s : ((lane == 18) ? s : 0.0f);
    bf.y = (lane == 1) ? -s : ((lane == 19) ? s : 0.0f);

    v8f cf = {};
    v8f d;
#if __has_builtin(__builtin_amdgcn_wmma_f32_16x16x4_f32)
    d = __builtin_amdgcn_wmma_f32_16x16x4_f32(false, af, false, bf,
                                              (short)0, cf, false, false);
#else
    // fragment-identical scalar fallback: D[row][n] = pred[n][row] * diag[n]
    {
        const float diag0 = -s, diag1 = -s, diag2 = s, diag3 = s;
        const float dg = (m == 0) ? diag0 : (m == 1) ? diag1 : (m == 2) ? diag2 : diag3;
        #pragma unroll
        for (int r = 0; r < 8; ++r) {
            const int row = r + (hi ? 8 : 0);
            float v = 0.0f;
            if (m < 4) {
                const int aa = a0 + row;
                v = (aa < LH)
                    ? pr[(size_t)img * NCH * (size_t)LH + (size_t)m * LH + aa] * dg
                    : 0.0f;
            }
            d[r] = v;
        }
    }
    (void)cf; (void)af; (void)bf;
#endif

    // Stage D fragment: VGPR r of lane t holds D[r + (t>=16?8:0)][t&15].
    if (m < 4) {
        #pragma unroll
        for (int r = 0; r < 8; ++r)
            shD[wave][m][r + (hi ? 8 : 0)] = d[r];
    }
    __syncthreads();

    const float dx1 = shD[wave][0][m];
    const float dy1 = shD[wave][1][m];
    const float dx2 = shD[wave][2][m];
    const float dy2 = shD[wave][3][m];

    const int   gx = a % nx, gy = a / nx;
    const float px = ((float)gx + 0.5f) * s;
    const float py = ((float)gy + 0.5f) * s;
    const float x1 = px + dx1, y1 = py + dy1;
    const float x2 = px + dx2, y2 = py + dy2;
    // reference round-trips cxcywh -> xyxy
    const float cx = 0.5f * (x1 + x2), cy = 0.5f * (y1 + y2);
    const float w  = x2 - x1,          h  = y2 - y1;
    const float bx1 = cx - 0.5f * w, by1 = cy - 0.5f * h;
    const float bx2 = cx + 0.5f * w, by2 = cy + 0.5f * h;

    // class max over logits; 40 classes per half-wave lane pair
    float mv = -FLT_MAX; int mi = 0;
    if (valid) {
        const int cstart = 5 + (hi ? 40 : 0);
        for (int c = 0; c < 40; ++c) {
            const float v = pr[base + (size_t)(cstart + c) * LH];
            if (v > mv) { mv = v; mi = cstart + c; }
        }
    }
    const float ov = __shfl_xor(mv, 16);
    const int   oi = __shfl_xor(mi, 16);
    if (ov > mv || (ov == mv && oi < mi)) { mv = ov; mi = oi; }

    const float objl    = valid ? pr[base + (size_t)4 * LH] : 0.0f;
    const float sobj    = sigm(objl);
    const float scls    = sigm(mv);
    const float confCol = sqrtf(sobj * scls);       // io[:,4]
    const float conf2   = scls * confCol;           // max cls_conf
    const float score   = (confCol > 0.2f && conf2 > 0.2f) ? conf2 : -1.0f;

    if (valid && !hi) {
        const int g = LOFF[lvl] + a;
        const size_t o = (size_t)img * ANCH_PAD + (size_t)g;
        float4 b4; b4.x = bx1; b4.y = by1; b4.z = bx2; b4.w = by2;
        ((float4*)wsBox)[o] = b4;
        wsCls[o]   = (float)(mi - 5);
        wsScore[o] = score;
    }
}

// ---------------------------------------------------------------------------
// Kernel 2: per-image bitonic top-K sort in 128KB LDS + greedy class-offset
// NMS + stable compaction to (300,6). One workgroup per image.
// ---------------------------------------------------------------------------
__global__ __launch_bounds__(1024) void nms_kernel(
    const float* __restrict__ wsBox, const float* __restrict__ wsCls,
    const float* __restrict__ wsScore, float* __restrict__ out)
{
    extern __shared__ unsigned char smem[];
    unsigned long long* keys = (unsigned long long*)smem;       // [16384] u64
    // overlay after sort: bytes >= 8192 == key slots >= 1024 (top-1000 intact)
    float* nbox  = (float*)(smem + 8192);                        // 1000*4 f32
    float* ncls  = (float*)(smem + 8192 + 16000);                // 1000 f32
    float* nsc   = (float*)(smem + 8192 + 20000);                // 1000 f32
    int*   nkeep = (int*)  (smem + 8192 + 24000);                // 1000 i32
    int*   nrank = (int*)  (smem + 8192 + 28000);                // 1000 i32

    const int img = blockIdx.x;
    const int tid = threadIdx.x;
    const int TB  = blockDim.x;

    // build sortable keys: order-preserving float bits | stable tie-break
    for (int j = tid; j < NSORT; j += TB) {
        const float sc = (j < NANCH) ? wsScore[(size_t)img * ANCH_PAD + j]
                                     : -FLT_MAX;
        unsigned u = __float_as_uint(sc);
        u = (u & 0x80000000u) ? ~u : (u | 0x80000000u);
        keys[j] = ((unsigned long long)u << 32) | (unsigned)(NSORT - 1 - j);
    }

    // descending bitonic sort of 16384 keys
    for (unsigned k = 2; k <= (unsigned)NSORT; k <<= 1) {
        for (unsigned jj = k >> 1; jj > 0; jj >>= 1) {
            __syncthreads();
            for (unsigned i = (unsigned)tid; i < (unsigned)NSORT; i += (unsigned)TB) {
                const unsigned ixj = i ^ jj;
                if (ixj > i) {
                    const unsigned long long a = keys[i], b = keys[ixj];
                    const bool desc = ((i & k) == 0);
                    if (desc ? (a < b) : (a > b)) { keys[i] = b; keys[ixj] = a; }
                }
            }
        }
    }
    __syncthreads();

    // gather top-1000 into overlay
    for (int j = tid; j < K_PRE; j += TB) {
        const unsigned long long key = keys[j];
        unsigned u = (unsigned)(key >> 32);
        const int idx = (NSORT - 1) - (int)((unsigned)key & 0xFFFFu);
        u = (u & 0x80000000u) ? (u & 0x7FFFFFFFu) : ~u;
        const float sc = __uint_as_float(u);
        const bool real = idx < NANCH;
        float4 b4;
        if (real) b4 = ((const float4*)wsBox)[(size_t)img * ANCH_PAD + idx];
        else      { b4.x = b4.y = b4.z = b4.w = 0.0f; }
        nbox[4 * j + 0] = b4.x; nbox[4 * j + 1] = b4.y;
        nbox[4 * j + 2] = b4.z; nbox[4 * j + 3] = b4.w;
        ncls[j]  = real ? wsCls[(size_t)img * ANCH_PAD + idx] : 0.0f;
        nsc[j]   = sc;
        nkeep[j] = (sc > 0.0f) ? 1 : 0;
    }

    // greedy sequential NMS (class-aware via +cls*4096 offsets, ref formula)
    for (int i = 0; i < K_PRE; ++i) {
        __syncthreads();
        if (!nkeep[i]) continue;
        const float offA = ncls[i] * 4096.0f;
        const float ax1 = nbox[4 * i + 0] + offA, ay1 = nbox[4 * i + 1] + offA;
        const float ax2 = nbox[4 * i + 2] + offA, ay2 = nbox[4 * i + 3] + offA;
        const float areaA = (ax2 - ax1) * (ay2 - ay1);
        for (int j = i + 1 + tid; j < K_PRE; j += TB) {
            if (!nkeep[j]) continue;
            const float offB = ncls[j] * 4096.0f;
            const float bx1 = nbox[4 * j + 0] + offB, by1 = nbox[4 * j + 1] + offB;
            const float bx2 = nbox[4 * j + 2] + offB, by2 = nbox[4 * j + 3] + offB;
            const float areaB = (bx2 - bx1) * (by2 - by1);
            const float ltx = fmaxf(ax1, bx1), lty = fmaxf(ay1, by1);
            const float rbx = fminf(ax2, bx2), rby = fminf(ay2, by2);
            const float iw = fmaxf(rbx - ltx, 0.0f), ih = fmaxf(rby - lty, 0.0f);
            const float inter = iw * ih;
            const float iou = inter / (areaA + areaB - inter + 1e-16f);
            if (iou > 0.6f) nkeep[j] = 0;
        }
    }
    __syncthreads();

    if (tid == 0) {
        int r = 0;
        for (int j = 0; j < K_PRE; ++j) { nrank[j] = r; if (nkeep[j]) r++; }
    }
    for (int j = tid; j < MAX_DET * 6; j += TB)
        out[(size_t)img * (MAX_DET * 6) + j] = 0.0f;
    __syncthreads();

    for (int j = tid; j < K_PRE; j += TB) {
        if (nkeep[j]) {
            const int r = nrank[j];
            if (r < MAX_DET) {
                float* o = out + (size_t)img * (MAX_DET * 6) + (size_t)r * 6;
                o[0] = nbox[4 * j + 0]; o[1] = nbox[4 * j + 1];
                o[2] = nbox[4 * j + 2]; o[3] = nbox[4 * j + 3];
                o[4] = nsc[j];          o[5] = ncls[j];
            }
        }
    }
}

extern "C" void kernel_launch(void* const* d_in, const int* in_sizes, int n_in,
                              void* d_out, int out_size, void* d_ws, size_t ws_size,
                              hipStream_t stream) {
    (void)in_sizes; (void)n_in; (void)out_size; (void)ws_size;
    // setup_inputs() interleaves: pred0,anchor0,pred1,anchor1,...
    const float* p0 = (const float*)d_in[0];
    const float* p1 = (const float*)d_in[2];
    const float* p2 = (const float*)d_in[4];
    const float* p3 = (const float*)d_in[6];
    const float* p4 = (const float*)d_in[8];

    float* wsBox   = (float*)d_ws;                                   // 32*13344*4
    float* wsCls   = wsBox + (size_t)NIMG * ANCH_PAD * 4;            // 32*13344
    float* wsScore = wsCls + (size_t)NIMG * ANCH_PAD;                // 32*13344
    float* out     = (float*)d_out;

    const int nTiles  = NIMG * TILES_PER_IMG;                        // 26784
    const int nBlocks = nTiles / WAVES_PER_BLK;                      // 3348 exact
    decode_kernel<<<nBlocks, 256, 0, stream>>>(p0, p1, p2, p3, p4,
                                               wsBox, wsCls, wsScore);

    const size_t smemBytes = (size_t)NSORT * sizeof(unsigned long long); // 128KB
    hipFuncSetAttribute((const void*)nms_kernel,
                        hipFuncAttributeMaxDynamicSharedMemorySize,
                        (int)smemBytes);
    nms_kernel<<<NIMG, 1024, smemBytes, stream>>>(wsBox, wsCls, wsScore, out);
}